// ParamEvaler_80745385165285
// MI455X (gfx1250) — compile-verified
//
#include <hip/hip_runtime.h>
#include <math.h>

// MI455X / gfx1250, wave32. Two kernels:
//  1) fused conv(5x5,valid) + bias + relu + 2x2 maxpool via V_WMMA_F32_16X16X4_F32
//     (implicit im2col from LDS), writes pooled [64][128][4608] f32 to workspace.
//     Weight A-fragments + biases are register-hoisted per init; each im2col
//     B-fragment feeds both channel-tiles (2 WMMAs per LDS gather pair).
//  2) dense (4608->10) via V_WMMA_F32_16X16X4_F32 + bias + log_softmax -> d_out.
// Workspace requirement: 64*128*4608*4 = 150,994,944 bytes.

typedef float v2f __attribute__((ext_vector_type(2)));
typedef float v8f __attribute__((ext_vector_type(8)));

#define PSTRIDE 46922   // floats per init in params
#define BC_OFF  800     // conv bias offset
#define WD_OFF  832     // dense weight offset
#define BD_OFF  46912   // dense bias offset

static __device__ __forceinline__ v8f wmma_f32x4(v2f a, v2f b, v8f c) {
  // 8 args: (neg_a, A, neg_b, B, c_mod, C, reuse_a, reuse_b)
  return __builtin_amdgcn_wmma_f32_16x16x4_f32(false, a, false, b, (short)0, c,
                                               false, false);
}

// ---------------------------------------------------------------------------
// Kernel 1: conv + bias + relu + 2x2 maxpool
// grid = (128 images, 4 init-chunks), block = 256 (8 waves)
// dynamic LDS: img 812 | w 896 | bias 32 | conv 18432  => 20172 floats = 80688 B
// ---------------------------------------------------------------------------
__global__ void conv_pool_kernel(const float* __restrict__ params,
                                 const float* __restrict__ batch,
                                 float* __restrict__ pooled) {
  extern __shared__ float smem[];
  float* img = smem;            // [29*28] (row 28 zeroed: harmless pad-tap reads)
  float* wl  = smem + 812;      // [32][28] weights, taps 25..27 zeroed
  float* bl  = smem + 1708;     // [32] conv bias
  float* cv  = smem + 1740;     // [32][576] conv output (post-relu)

  const int n    = blockIdx.x;          // image index 0..127
  const int tid  = threadIdx.x;
  const int lane = tid & 31;
  const int wave = tid >> 5;
  const int col  = lane & 15;
  const int h    = lane >> 4;           // lane half: K offset 2*h

  // stage image once (784 real + zero pad row)
  for (int idx = tid; idx < 812; idx += 256)
    img[idx] = (idx < 784) ? batch[n * 784 + idx] : 0.0f;

  for (int ii = 0; ii < 16; ++ii) {
    const int i = blockIdx.y * 16 + ii;
    const float* P = params + (size_t)i * PSTRIDE;

    __syncthreads();  // prev iter's pool reads done before wl overwrite; img ready
    for (int idx = tid; idx < 896; idx += 256) {
      int c = idx / 28, t = idx - c * 28;
      wl[idx] = (t < 25) ? P[c * 25 + t] : 0.0f;   // zero-pad K to 28
    }
    if (tid < 32) bl[tid] = P[BC_OFF + tid];
    __syncthreads();

    // Hoist A fragments (weights) for both channel-tiles: invariant over pt.
    // Layout: A element (M=lane%16, K = 4*s + 2*h + vgpr)
    v2f aw0[7], aw1[7];
#pragma unroll
    for (int s = 0; s < 7; ++s) {
      aw0[s] = *(const v2f*)&wl[(col)      * 28 + 4 * s + 2 * h];
      aw1[s] = *(const v2f*)&wl[(16 + col) * 28 + 4 * s + 2 * h];
    }
    // Hoist conv biases for the D rows this lane produces (M = r + 8h).
    float bv0[8], bv1[8];
#pragma unroll
    for (int r = 0; r < 8; ++r) {
      bv0[r] = bl[r + 8 * h];
      bv1[r] = bl[16 + r + 8 * h];
    }

    // 36 position-tiles; each wave does both channel-tiles per position-tile
    // so every im2col gather feeds two WMMAs.
    for (int pt = wave; pt < 36; pt += 8) {
      const int p  = pt * 16 + col;       // conv output position 0..575
      const int y  = p / 24;
      const int x  = p - y * 24;
      const int ib = y * 28 + x;

      v8f c0a = {}, c0b = {}, c1a = {}, c1b = {};   // 4 independent chains
#pragma unroll
      for (int s = 0; s < 7; ++s) {
        // B: implicit im2col, taps t0,t0+1 at position p (N = lane%16)
        int t0  = 4 * s + 2 * h;
        int dy0 = t0 / 5,  dx0 = t0 - 5 * dy0;
        int t1  = t0 + 1;
        int dy1 = t1 / 5,  dx1 = t1 - 5 * dy1;
        v2f b;
        b.x = img[ib + dy0 * 28 + dx0];
        b.y = img[ib + dy1 * 28 + dx1];
        if (s & 1) {
          c0b = wmma_f32x4(aw0[s], b, c0b);
          c1b = wmma_f32x4(aw1[s], b, c1b);
        } else {
          c0a = wmma_f32x4(aw0[s], b, c0a);
          c1a = wmma_f32x4(aw1[s], b, c1a);
        }
      }
      // D: M = r + 8h -> channel; N = col -> position. bias + relu -> LDS
#pragma unroll
      for (int r = 0; r < 8; ++r) {
        int c0 = r + 8 * h;
        float v0 = c0a[r] + c0b[r] + bv0[r];
        cv[c0 * 576 + p] = v0 > 0.0f ? v0 : 0.0f;
        int c1 = 16 + r + 8 * h;
        float v1 = c1a[r] + c1b[r] + bv1[r];
        cv[c1 * 576 + p] = v1 > 0.0f ? v1 : 0.0f;
      }
    }
    __syncthreads();

    // 2x2 max-pool -> pooled[i][n][c*144 + py*12 + px]
    float* outp = pooled + ((size_t)i * 128 + n) * 4608;
    for (int idx = tid; idx < 4608; idx += 256) {
      int c   = idx / 144;
      int rem = idx - c * 144;
      int py  = rem / 12;
      int px  = rem - py * 12;
      const float* rp = &cv[c * 576 + (2 * py) * 24 + 2 * px];
      float m0 = fmaxf(rp[0],  rp[1]);
      float m1 = fmaxf(rp[24], rp[25]);
      outp[idx] = fmaxf(m0, m1);
    }
  }
}

// ---------------------------------------------------------------------------
// Kernel 2: dense (M=16 images, N=16 outs [10 valid], K=4608) + log_softmax
// grid = 64 inits * 8 image-tiles = 512 blocks, block = 32 (one wave)
// ---------------------------------------------------------------------------
__global__ void dense_softmax_kernel(const float* __restrict__ params,
                                     const float* __restrict__ pooled,
                                     float* __restrict__ out) {
  __shared__ float tile[16 * 17];

  const int b    = blockIdx.x;
  const int i    = b >> 3;
  const int m0   = (b & 7) * 16;
  const int lane = threadIdx.x & 31;
  const int col  = lane & 15;
  const int h    = lane >> 4;

  const float* P = params + (size_t)i * PSTRIDE;
  const float* Arow = pooled + ((size_t)i * 128 + m0 + col) * 4608 + 2 * h;
  const bool valid = (col < 10);
  // clamp invalid lanes to row 0 (avoids OOB on last init); cols 10..15 of D
  // become copies of col 0 and are never read.
  const float* Brow = P + WD_OFF + (size_t)(valid ? col : 0) * 4608 + 2 * h;

  v8f a0c = {}; v8f a1c = {}; v8f a2c = {}; v8f a3c = {};
  for (int kk = 0; kk < 4608; kk += 16) {
    v2f aa0 = *(const v2f*)(Arow + kk);
    v2f bb0 = *(const v2f*)(Brow + kk);
    v2f aa1 = *(const v2f*)(Arow + kk + 4);
    v2f bb1 = *(const v2f*)(Brow + kk + 4);
    v2f aa2 = *(const v2f*)(Arow + kk + 8);
    v2f bb2 = *(const v2f*)(Brow + kk + 8);
    v2f aa3 = *(const v2f*)(Arow + kk + 12);
    v2f bb3 = *(const v2f*)(Brow + kk + 12);
    a0c = wmma_f32x4(aa0, bb0, a0c);
    a1c = wmma_f32x4(aa1, bb1, a1c);
    a2c = wmma_f32x4(aa2, bb2, a2c);
    a3c = wmma_f32x4(aa3, bb3, a3c);
  }
  v8f acc = (a0c + a1c) + (a2c + a3c);

  const float bias = valid ? P[BD_OFF + col] : 0.0f;
#pragma unroll
  for (int r = 0; r < 8; ++r)
    tile[(r + 8 * h) * 17 + col] = acc[r] + bias;   // D: M=r+8h, N=col
  __syncthreads();

  if (lane < 16) {
    const int m = lane;
    float mx = -1e30f;
#pragma unroll
    for (int o = 0; o < 10; ++o) mx = fmaxf(mx, tile[m * 17 + o]);
    float s = 0.0f;
#pragma unroll
    for (int o = 0; o < 10; ++o) s += expf(tile[m * 17 + o] - mx);
    const float lse = mx + logf(s);
    float* op = out + ((size_t)i * 128 + m0 + m) * 10;
#pragma unroll
    for (int o = 0; o < 10; ++o) op[o] = tile[m * 17 + o] - lse;
  }
}

// ---------------------------------------------------------------------------
extern "C" void kernel_launch(void* const* d_in, const int* in_sizes, int n_in,
                              void* d_out, int out_size, void* d_ws, size_t ws_size,
                              hipStream_t stream) {
  (void)in_sizes; (void)n_in; (void)out_size; (void)ws_size;
  const float* params = (const float*)d_in[0];  // [64, 46922]
  const float* batch  = (const float*)d_in[1];  // [128, 1, 28, 28]
  float* out    = (float*)d_out;                // [64, 128, 10]
  float* pooled = (float*)d_ws;                 // [64, 128, 4608] f32

  dim3 g1(128, 4);
  conv_pool_kernel<<<g1, 256, 80688, stream>>>(params, batch, pooled);
  dense_softmax_kernel<<<512, 32, 0, stream>>>(params, pooled, out);
}